// BoundaryLoss_87540023427570
// MI455X (gfx1250) — compile-verified
//
#include <hip/hip_runtime.h>
#include <hip/hip_bf16.h>
#include <stdint.h>

// Problem constants from the reference (fixed shapes).
#define BATCH 8
#define H 256
#define W 256
#define NPIX (BATCH * H * W)      // 524288
#define NROWS (BATCH * H)         // 2048
#define NCOLS (BATCH * W)         // 2048
#define EDT_INF 1000000.0f

typedef float v2f __attribute__((ext_vector_type(2)));
typedef float v8f __attribute__((ext_vector_type(8)));

// ---------------------------------------------------------------------------
// Kernel A: zero the per-batch has_pos/has_neg flags (atomicOr targets).
// ---------------------------------------------------------------------------
__global__ void sdt_init_kernel(int* flags) {
    if (threadIdx.x < BATCH) flags[threadIdx.x] = 0;
}

// ---------------------------------------------------------------------------
// Kernel B: per-row 1D EDT (forward + backward scan), squared & clamped.
// One thread per image row. Mask bits live in 8 u32 registers (no spills).
//   g2N = min(g_to_nearest_NEG, INF)^2   (feeds pos_edt = EDT of background)
//   g2P = min(g_to_nearest_POS, INF)^2   (feeds neg_edt = EDT of foreground)
// Also ORs per-batch "any pos / any neg" flags (idempotent -> deterministic).
// ---------------------------------------------------------------------------
__global__ __launch_bounds__(256)
void sdt_rows_kernel(const float* __restrict__ seg,
                     float* __restrict__ g2N,
                     float* __restrict__ g2P,
                     int* __restrict__ flags) {
    const int t = blockIdx.x * blockDim.x + threadIdx.x;   // row id in [0, NROWS)
    const int b = t >> 8;
    const int y = t & 255;
    const long rowBase = (long)t * W;
    const long seg1Base = (((long)b * 3 + 1) * H + y) * W;
    const long seg2Base = (((long)b * 3 + 2) * H + y) * W;

    // Build packed pos-mask for this row.
    unsigned mpos[8];
    bool anyPos = false, anyNeg = false;
#pragma unroll
    for (int w = 0; w < 8; ++w) {
        unsigned word = 0;
        for (int i = 0; i < 32; ++i) {
            const int x = w * 32 + i;
            const float bin = seg[seg1Base + x] + seg[seg2Base + x];
            if (bin > 0.5f) word |= (1u << i);
        }
        mpos[w] = word;
        anyPos |= (word != 0u);
        anyNeg |= (word != 0xffffffffu);
    }

    // Forward scan: d = bg ? 0 : d_prev + 1, init INF (matches lax.scan).
    float dN = EDT_INF, dP = EDT_INF;
#pragma unroll
    for (int w = 0; w < 8; ++w) {
        const unsigned word = mpos[w];
        for (int i = 0; i < 32; ++i) {
            const int x = w * 32 + i;
            const bool p = (word >> i) & 1u;
            dN = p ? (dN + 1.0f) : 0.0f;   // bg for N-transform is "neg" (!p)
            dP = p ? 0.0f : (dP + 1.0f);   // bg for P-transform is "pos" (p)
            g2N[rowBase + x] = dN;          // stash forward pass
            g2P[rowBase + x] = dP;
        }
    }

    // Backward scan: combine, clamp to INF, square in-place.
    dN = EDT_INF; dP = EDT_INF;
#pragma unroll
    for (int w = 7; w >= 0; --w) {
        const unsigned word = mpos[w];
        for (int i = 31; i >= 0; --i) {
            const int x = w * 32 + i;
            const bool p = (word >> i) & 1u;
            dN = p ? (dN + 1.0f) : 0.0f;
            dP = p ? 0.0f : (dP + 1.0f);
            float gN = fminf(fminf(g2N[rowBase + x], dN), EDT_INF);
            float gP = fminf(fminf(g2P[rowBase + x], dP), EDT_INF);
            g2N[rowBase + x] = gN * gN;
            g2P[rowBase + x] = gP * gP;
        }
    }

    int f = (anyPos ? 1 : 0) | (anyNeg ? 2 : 0);
    if (f) atomicOr(&flags[b], f);
}

// ---------------------------------------------------------------------------
// Wave32 sum via V_WMMA_F32_16X16X4_F32 (tensor-core reduction).
// A: lane's value at (M = lane%16, K = lane<16 ? 0 : 2), second VGPR zero.
// B: all ones -> D[m][n] = v[m] + v[m+16] for every n.
// Per-lane sum of the 8 D VGPRs gives half-wave sums; xor-shuffle by 16
// combines them into the full 32-lane sum. Requires EXEC all ones (holds:
// grid sizes divide exactly, no divergence at call sites).
// ---------------------------------------------------------------------------
__device__ __forceinline__ float wave32_sum_wmma(float v) {
    v2f a; a.x = v;    a.y = 0.0f;
    v2f bm; bm.x = 1.0f; bm.y = 1.0f;
    v8f c = {};
    v8f d = __builtin_amdgcn_wmma_f32_16x16x4_f32(
        /*neg_a=*/false, a, /*neg_b=*/false, bm,
        /*c_mod=*/(short)0, c, /*reuse_a=*/false, /*reuse_b=*/false);
    float s = d[0] + d[1] + d[2] + d[3] + d[4] + d[5] + d[6] + d[7];
    s += __shfl_xor(s, 16, 32);   // wave32: combine the two half-wave sums
    return s;
}

// ---------------------------------------------------------------------------
// Kernel C: per-column squared lower envelope (exact 2D EDT), fused with
// signed-distance assembly + weighted-L1 loss + deterministic block reduce.
// One block per (b, x) column; thread y owns one output pixel. The column
// gather (1KB stride, L2-resident) is staged memory->LDS with the CDNA5
// async mover (GLOBAL_LOAD_ASYNC_TO_LDS_B32, tracked by ASYNCcnt): no VGPR
// round-trip, one s_wait_asynccnt per wave before the barrier.
// ---------------------------------------------------------------------------
__global__ __launch_bounds__(256)
void sdt_cols_loss_kernel(const float* __restrict__ g2N,
                          const float* __restrict__ g2P,
                          const float* __restrict__ pred,
                          const int* __restrict__ flags,
                          float* __restrict__ partials) {
    __shared__ float sgN[H];
    __shared__ float sgP[H];
    __shared__ float wsum[8];

    const int x = blockIdx.x & (W - 1);
    const int b = blockIdx.x >> 8;
    const int y = threadIdx.x;

    // Async-stage this column of both g^2 fields into LDS.
    // VDST operand = per-lane LDS byte offset (generic ptr low 32 bits map to
    // the LDS aperture offset); VADDR = per-lane 64-bit global address.
    const long colBase = ((long)b * H) * W + x;
    const float* srcN = g2N + colBase + (long)y * W;
    const float* srcP = g2P + colBase + (long)y * W;
    const unsigned ldsN = (unsigned)(uintptr_t)&sgN[y];
    const unsigned ldsP = (unsigned)(uintptr_t)&sgP[y];
    asm volatile("global_load_async_to_lds_b32 %0, %1, off"
                 :: "v"(ldsN), "v"(srcN) : "memory");
    asm volatile("global_load_async_to_lds_b32 %0, %1, off"
                 :: "v"(ldsP), "v"(srcP) : "memory");

    // Prefetch the pred pixel we consume after the envelope loop.
    const float* predPtr = pred + ((long)b * H + y) * W + x;
    __builtin_prefetch(predPtr, 0, 0);

    asm volatile("s_wait_asynccnt 0x0" ::: "memory");  // wave's LDS writes done
    __syncthreads();                                    // visible block-wide

    // Lower envelope: d2 = min_{y'} (y - y')^2 + g2[y'].
    const float fy = (float)y;
    float mN = 1e30f, mP = 1e30f;
#pragma unroll 8
    for (int yp = 0; yp < H; ++yp) {
        const float t  = fy - (float)yp;
        const float t2 = t * t;                 // exact in f32 (<= 65025)
        mN = fminf(mN, t2 + sgN[yp]);
        mP = fminf(mP, t2 + sgP[yp]);
    }

    const int f = flags[b];
    const float dPos = (f & 1) ? sqrtf(mN) : 0.0f;  // pos_edt = EDT of neg
    const float dNeg = (f & 2) ? sqrtf(mP) : 0.0f;  // neg_edt = EDT of pos
    const float sdt  = dPos - dNeg;

    const float p    = *predPtr;
    const float loss = fabsf(p - sdt) * expf(-fabsf(sdt) * 0.2f);

    // Deterministic block sum: WMMA wave reduction, then fixed-order combine.
    const float s = wave32_sum_wmma(loss);
    const int lane = threadIdx.x & 31;
    const int wave = threadIdx.x >> 5;
    if (lane == 0) wsum[wave] = s;
    __syncthreads();
    if (threadIdx.x == 0) {
        float tot = 0.0f;
        for (int i = 0; i < 8; ++i) tot += wsum[i];
        partials[blockIdx.x] = tot;
    }
}

// ---------------------------------------------------------------------------
// Kernel D: deterministic final reduction of the 2048 block partials -> mean.
// Fixed-order per-thread strided sums, WMMA wave reduction, fixed-order
// cross-wave combine (bit-stable across replays).
// ---------------------------------------------------------------------------
__global__ __launch_bounds__(256)
void sdt_final_kernel(const float* __restrict__ partials, float* __restrict__ out) {
    __shared__ float wsum[8];
    float s = 0.0f;
    for (int i = threadIdx.x; i < NCOLS; i += 256) s += partials[i];

    const float ws = wave32_sum_wmma(s);
    const int lane = threadIdx.x & 31;
    const int wave = threadIdx.x >> 5;
    if (lane == 0) wsum[wave] = ws;
    __syncthreads();
    if (threadIdx.x == 0) {
        float tot = 0.0f;
        for (int i = 0; i < 8; ++i) tot += wsum[i];
        out[0] = tot * (1.0f / (float)NPIX);
    }
}

// ---------------------------------------------------------------------------
// Host launcher. ws layout (floats): [0,NPIX) g2N | [NPIX,2*NPIX) g2P |
// [2*NPIX, 2*NPIX+NCOLS) partials | then 8 ints of flags. ~4.2 MB total.
// ---------------------------------------------------------------------------
extern "C" void kernel_launch(void* const* d_in, const int* in_sizes, int n_in,
                              void* d_out, int out_size, void* d_ws, size_t ws_size,
                              hipStream_t stream) {
    const float* pred = (const float*)d_in[0];   // (8,1,256,256) f32
    const float* seg  = (const float*)d_in[1];   // (8,3,256,256) f32

    float* g2N      = (float*)d_ws;
    float* g2P      = g2N + NPIX;
    float* partials = g2P + NPIX;
    int*   flags    = (int*)(partials + NCOLS);

    hipLaunchKernelGGL(sdt_init_kernel, dim3(1), dim3(32), 0, stream, flags);
    hipLaunchKernelGGL(sdt_rows_kernel, dim3(NROWS / 256), dim3(256), 0, stream,
                       seg, g2N, g2P, flags);
    hipLaunchKernelGGL(sdt_cols_loss_kernel, dim3(NCOLS), dim3(256), 0, stream,
                       g2N, g2P, pred, flags, partials);
    hipLaunchKernelGGL(sdt_final_kernel, dim3(1), dim3(256), 0, stream,
                       partials, (float*)d_out);
}